// BiLingual_44341242364622
// MI455X (gfx1250) — compile-verified
//
#include <hip/hip_runtime.h>
#include <stdint.h>

// Problem constants (from reference): B=64, S=2048, V=50000, D=256
constexpr int kB = 64;
constexpr int kS = 2048;
constexpr int kD = 256;

typedef __attribute__((ext_vector_type(2))) float v2f;
typedef __attribute__((ext_vector_type(8))) float v8f;

__device__ __forceinline__ float fast_tanh(float x) {
#if __has_builtin(__builtin_amdgcn_tanhf)
  return __builtin_amdgcn_tanhf(x);
#else
  float y;
  asm("v_tanh_f32 %0, %1" : "=v"(y) : "v"(x));
  return y;
#endif
}

// Load with uniform base + 32-bit byte offset (lowers to global_load saddr form).
__device__ __forceinline__ float ldg_boff(const float* base, uint32_t byte_off) {
  return *(const float*)((const char*)base + byte_off);
}

// One block = (table t, batch b, sequence chunk). 512 threads = 16 waves.
// Wave w owns channels [16w, 16w+16); both lane halves of a wave compute the
// SAME 16 channels but walk different halves of the chunk's pair range, feeding
// K={0,1} (lanes 0-15) and K={2,3} (lanes 16-31) of V_WMMA_F32_16X16X4_F32 with
// A = all-ones 16x4, so D[m][n] = sum_k B[k][n] + C[m][n]: chaining D->C
// accumulates per-channel sums on the matrix pipe while the VALU does add+tanh.
// NO guards: pad slots (staged vocab index 0) are subtracted in finalize().
extern "C" __global__ void __launch_bounds__(512)
bilingual_addtanh_pool(const int* __restrict__ idx_pri,
                       const int* __restrict__ idx_sec,
                       const float* __restrict__ W_pri,
                       const float* __restrict__ W_sec,
                       float* __restrict__ dst,
                       int chunks, int P) {
  extern __shared__ int lds_off[];      // staged byte offsets: idx[g] * kD * 4
  const int chunk = blockIdx.x;
  const int b     = blockIdx.y;
  const int t     = blockIdx.z;
  const int tid   = threadIdx.x;
  const int wave  = tid >> 5;
  const int lane  = tid & 31;
  const int n     = lane & 15;
  const int khalf = lane >> 4;
  const uint32_t c4 = (uint32_t)(wave * 16 + n) * sizeof(float); // channel byte off

  const int*   idx = (t == 0 ? idx_pri : idx_sec) + b * kS;
  const float* W   = (t == 0 ? W_pri   : W_sec);

  // Stage rows [p0, p0+P] as byte offsets; rows past the sequence end -> row 0.
  const int p0 = chunk * P;
  for (int r = tid; r <= P; r += blockDim.x) {
    const int g = p0 + r;
    lds_off[r] = ((g < kS) ? idx[g] : 0) * (kD * (int)sizeof(float));
  }
  __syncthreads();

  const v2f ones = {1.0f, 1.0f};
  v8f acc0 = {}, acc1 = {};

  const int half = P >> 1;              // pairs per lane-half (even, P % 8 == 0)
  const int base = khalf * half;        // this lane-half's first local pair slot

  float eprev = ldg_boff(W, (uint32_t)lds_off[base] + c4);
#pragma unroll 2
  for (int q = 0; q < half; q += 4) {
    const int o1 = lds_off[base + q + 1];
    const int o2 = lds_off[base + q + 2];
    const int o3 = lds_off[base + q + 3];
    const int o4 = lds_off[base + q + 4];
    const float e1 = ldg_boff(W, (uint32_t)o1 + c4);
    const float e2 = ldg_boff(W, (uint32_t)o2 + c4);
    const float e3 = ldg_boff(W, (uint32_t)o3 + c4);
    const float e4 = ldg_boff(W, (uint32_t)o4 + c4);
    v2f bmA = {fast_tanh(eprev + e1), fast_tanh(e1 + e2)};
    v2f bmB = {fast_tanh(e2 + e3),    fast_tanh(e3 + e4)};
    acc0 = __builtin_amdgcn_wmma_f32_16x16x4_f32(false, ones, false, bmA,
                                                 (short)0, acc0, false, false);
    acc1 = __builtin_amdgcn_wmma_f32_16x16x4_f32(false, ones, false, bmB,
                                                 (short)0, acc1, false, false);
    eprev = e4;
  }

  const v8f acc = acc0 + acc1;          // every D row holds the column sums
  if (lane < 16) {
    dst[((size_t)(t * kB + b) * chunks + chunk) * kD + (wave * 16 + n)] = acc[0];
  }
}

// Reduce chunk partials (or pass through the direct-write result) and subtract
// the analytically-known pad contributions:
//   1 x tanh(W[idx[b][S-1]][d] + W[0][d])  +  (n_bogus-1) x tanh(2*W[0][d]).
extern "C" __global__ void finalize(const float* __restrict__ part,
                                    float* __restrict__ out,
                                    const int* __restrict__ idx_pri,
                                    const int* __restrict__ idx_sec,
                                    const float* __restrict__ W_pri,
                                    const float* __restrict__ W_sec,
                                    int chunks, int n_bogus) {
  const int o = blockIdx.x * blockDim.x + threadIdx.x;
  if (o >= 2 * kB * kD) return;
  const int tb = o / kD;
  const int d  = o % kD;
  const int t  = tb / kB;
  const int b  = tb % kB;
  const int*   idx = (t == 0) ? idx_pri : idx_sec;
  const float* W   = (t == 0) ? W_pri   : W_sec;

  float s;
  if (chunks >= 2) {
    s = 0.0f;
    for (int ck = 0; ck < chunks; ++ck)
      s += part[((size_t)tb * chunks + ck) * kD + d];
  } else {
    s = out[o];                          // direct-write path (main wrote out)
  }
  const float w0 = W[d];                 // vocab row 0
  const int  il  = idx[b * kS + (kS - 1)];
  s -= fast_tanh(W[(size_t)il * kD + d] + w0);
  s -= (float)(n_bogus - 1) * fast_tanh(w0 + w0);
  out[o] = s;
}

extern "C" void kernel_launch(void* const* d_in, const int* in_sizes, int n_in,
                              void* d_out, int out_size, void* d_ws, size_t ws_size,
                              hipStream_t stream) {
  const int*   idx_pri = (const int*)d_in[0];
  const int*   idx_sec = (const int*)d_in[1];
  const float* W_pri   = (const float*)d_in[2];
  const float* W_sec   = (const float*)d_in[3];
  float*       out     = (float*)d_out;

  // Partial buffer: [2*B, chunks, D] f32 in d_ws; one "layer" = 128 KiB.
  const size_t layer_bytes = (size_t)2 * kB * kD * sizeof(float);
  const size_t fit = ws_size / layer_bytes;
  const int chunks = (fit >= 2) ? (int)(fit < 16 ? fit : 16) : 1;

  // Pair slots per chunk: multiple of 8 so each lane-half walks an even count.
  const int P = (((kS - 1) + chunks - 1) / chunks + 7) & ~7;
  const int n_bogus = chunks * P - (kS - 1);       // always >= 1 (chunks*P even)
  const size_t smem = (size_t)(P + 2) * sizeof(int);

  float* dst = (chunks >= 2) ? (float*)d_ws : out;
  dim3 grid(chunks, kB, 2);
  bilingual_addtanh_pool<<<grid, 512, smem, stream>>>(
      idx_pri, idx_sec, W_pri, W_sec, dst, chunks, P);

  const int nout = 2 * kB * kD;
  finalize<<<(nout + 255) / 256, 256, 0, stream>>>(
      (const float*)d_ws, out, idx_pri, idx_sec, W_pri, W_sec, chunks, n_bogus);
}